// GNNConv_32117765440104
// MI455X (gfx1250) — compile-verified
//
#include <hip/hip_runtime.h>
#include <math.h>

typedef __attribute__((ext_vector_type(2))) float v2f;
typedef __attribute__((ext_vector_type(8))) float v8f;

#define ST_STRIDE 68          // 16x64 tile staged with padded stride -> no LDS bank conflicts
#define WAVES_PER_BLOCK 8

// ---------- math helpers (match jax.nn.softplus / mish in fp32) ----------
__device__ __forceinline__ float softplus_f(float x) {
    // logaddexp(x, 0) stable form
    return fmaxf(x, 0.0f) + log1pf(expf(-fabsf(x)));
}
__device__ __forceinline__ float mish_f(float x) {
    return x * tanhf(softplus_f(x));
}

// ---------- one Linear(64,64): D(16x64) = X(16x64) @ W^T, via 16 k-steps x 4 n-tiles ----------
// st: LDS tile, row-major with ST_STRIDE padding. pk: packed B-fragments (4096 floats).
// Fragment packing: pk[ks*256 + t*64 + lane*2 + v] = W[n*64 + k],
//   n = 16*t + (lane&15), k = 4*ks + 2*(lane>>4) + v   (B[k][n] = W^T)
__device__ __forceinline__ void gemm16x64_layer(const float* st, const float* pk,
                                                v8f acc[4], int lane) {
    const int m  = lane & 15;
    const int hi = lane >> 4;
    for (int ks = 0; ks < 16; ++ks) {
        // A fragment: a[v] = X[m][4*ks + 2*hi + v]  (8B aligned, conflict-free banks)
        v2f a = *(const v2f*)(st + m * ST_STRIDE + ks * 4 + 2 * hi);
        const float* bp = pk + ks * 256 + lane * 2;
#pragma unroll
        for (int t = 0; t < 4; ++t) {
            v2f b = *(const v2f*)(bp + t * 64);
            acc[t] = __builtin_amdgcn_wmma_f32_16x16x4_f32(
                false, a, false, b, (short)0, acc[t], false, false);
        }
    }
}

// ---------- weight fragment packer: 8 layers x 4096 elems ----------
__global__ __launch_bounds__(256) void pack_weights_kernel(
    const float* __restrict__ w0, const float* __restrict__ w1,
    const float* __restrict__ w2, const float* __restrict__ w3,
    const float* __restrict__ w4, const float* __restrict__ w5,
    const float* __restrict__ w6, const float* __restrict__ w7,
    float* __restrict__ packed) {
    const float* Ws[8] = {w0, w1, w2, w3, w4, w5, w6, w7};
    const int layer = blockIdx.y;
    const float* W = Ws[layer];
    const int tid = blockIdx.x * blockDim.x + threadIdx.x;   // 0..4095
    const int ks  = tid >> 8;
    const int rem = tid & 255;
    const int t   = rem >> 6;
    const int l   = (rem >> 1) & 31;
    const int v   = rem & 1;
    const int n   = 16 * t + (l & 15);
    const int k   = 4 * ks + 2 * (l >> 4) + v;
    packed[(size_t)layer * 4096 + tid] = W[n * 64 + k];
}

// ---------- node MLP: out[tile rows] = Linear2(mish(Linear1(X))) ----------
__global__ __launch_bounds__(256) void node_mlp_kernel(
    const float* __restrict__ X,
    const float* __restrict__ pk1, const float* __restrict__ pk2,
    const float* __restrict__ b1,  const float* __restrict__ b2,
    float* __restrict__ out, int nrows) {
    __shared__ __align__(16) float stage[WAVES_PER_BLOCK][16 * ST_STRIDE];
    const int wave = threadIdx.x >> 5;
    const int lane = threadIdx.x & 31;
    const int tile = blockIdx.x * WAVES_PER_BLOCK + wave;
    const long base = (long)tile * 16;
    if (base >= nrows) return;                 // wave-uniform: EXEC stays all-ones
    float* st = stage[wave];

    const float* Xg = X + base * 64;
    const int valid = (nrows - base >= 16) ? 1024 : (int)(nrows - base) * 64;
    for (int i = lane; i < 1024; i += 32) {    // coalesced global -> padded LDS
        int ii = (i < valid) ? i : 0;
        st[(i >> 6) * ST_STRIDE + (i & 63)] = Xg[ii];
    }

    const int hi = lane >> 4, nb = lane & 15;
    v8f acc[4] = {};
    gemm16x64_layer(st, pk1, acc, lane);
#pragma unroll
    for (int t = 0; t < 4; ++t) {
        float bb = b1[16 * t + nb];
#pragma unroll
        for (int v = 0; v < 8; ++v)
            st[(v + 8 * hi) * ST_STRIDE + 16 * t + nb] = mish_f(acc[t][v] + bb);
    }
    v8f acc2[4] = {};
    gemm16x64_layer(st, pk2, acc2, lane);

    float* Og = out + base * 64;
#pragma unroll
    for (int t = 0; t < 4; ++t) {
        float bb = b2[16 * t + nb];
#pragma unroll
        for (int v = 0; v < 8; ++v) {
            int m2 = v + 8 * hi;
            if (base + m2 < nrows) Og[m2 * 64 + 16 * t + nb] = acc2[t][v] + bb;
        }
    }
}

// ---------- fused edge pipeline: ef = MLP_we(E); h = hs*hd*e^2/(4*pi*eps0*ef*1e-10);
//            m = MLP_wm(h); atomic scatter-add into acc[dst] ----------
__global__ __launch_bounds__(256) void edge_pipeline_kernel(
    const float* __restrict__ EF, const int* __restrict__ src, const int* __restrict__ dst,
    const float* __restrict__ hws, const float* __restrict__ hwd,
    const float* __restrict__ pk_e1, const float* __restrict__ pk_e2,
    const float* __restrict__ be1,   const float* __restrict__ be2,
    const float* __restrict__ pk_m1, const float* __restrict__ pk_m2,
    const float* __restrict__ bm1,   const float* __restrict__ bm2,
    float* __restrict__ acc_out, int nedges) {
    __shared__ __align__(16) float stage[WAVES_PER_BLOCK][16 * ST_STRIDE];
    const int wave = threadIdx.x >> 5;
    const int lane = threadIdx.x & 31;
    const int tile = blockIdx.x * WAVES_PER_BLOCK + wave;
    const long base = (long)tile * 16;
    if (base >= nedges) return;
    float* st = stage[wave];

    const float C_E   = 1.602176634e-19f;
    const float C_NUM = C_E * C_E;                       // fp32, like reference
    const float C_4PI_EPS0 = 12.566371f * 8.8541878128e-12f;

    const float* Eg = EF + base * 64;
    const int valid = (nedges - base >= 16) ? 1024 : (int)(nedges - base) * 64;
    for (int i = lane; i < 1024; i += 32) {
        int ii = (i < valid) ? i : 0;
        st[(i >> 6) * ST_STRIDE + (i & 63)] = Eg[ii];
    }

    const int hi = lane >> 4, nb = lane & 15;

    // edge MLP layer 1 + mish
    v8f a1[4] = {};
    gemm16x64_layer(st, pk_e1, a1, lane);
#pragma unroll
    for (int t = 0; t < 4; ++t) {
        float bb = be1[16 * t + nb];
#pragma unroll
        for (int v = 0; v < 8; ++v)
            st[(v + 8 * hi) * ST_STRIDE + 16 * t + nb] = mish_f(a1[t][v] + bb);
    }
    // edge MLP layer 2 -> ef (kept in D-layout registers)
    v8f ef[4] = {};
    gemm16x64_layer(st, pk_e2, ef, lane);

    // gather indices once per lane (row depends only on v,hi)
    int sidx[8], didx[8];
#pragma unroll
    for (int v = 0; v < 8; ++v) {
        long row = base + v + 8 * hi;
        if (row >= nedges) row = nedges - 1;
        sidx[v] = src[row];
        didx[v] = dst[row];
    }

    // Coulomb elementwise in D-layout, restage h_nodes into LDS
#pragma unroll
    for (int t = 0; t < 4; ++t) {
        float bb = be2[16 * t + nb];
        int n = 16 * t + nb;
#pragma unroll
        for (int v = 0; v < 8; ++v) {
            float e = ef[t][v] + bb;
            float hs = hws[(size_t)sidx[v] * 64 + n];
            float hd = hwd[(size_t)didx[v] * 64 + n];
            float num = hs * hd * C_NUM;
            float den = (C_4PI_EPS0 * e) * 1e-10f;       // same op order as reference
            st[(v + 8 * hi) * ST_STRIDE + n] = num / den;
        }
    }

    // message MLP
    v8f m1[4] = {};
    gemm16x64_layer(st, pk_m1, m1, lane);
#pragma unroll
    for (int t = 0; t < 4; ++t) {
        float bb = bm1[16 * t + nb];
#pragma unroll
        for (int v = 0; v < 8; ++v)
            st[(v + 8 * hi) * ST_STRIDE + 16 * t + nb] = mish_f(m1[t][v] + bb);
    }
    v8f m2[4] = {};
    gemm16x64_layer(st, pk_m2, m2, lane);

    // scatter-add messages into destination nodes
#pragma unroll
    for (int t = 0; t < 4; ++t) {
        float bb = bm2[16 * t + nb];
        int n = 16 * t + nb;
#pragma unroll
        for (int v = 0; v < 8; ++v) {
            if (base + v + 8 * hi < nedges)
                atomicAdd(&acc_out[(size_t)didx[v] * 64 + n], m2[t][v] + bb);
        }
    }
}

// ---------- final: out = softplus(node_feats + acc) ----------
__global__ __launch_bounds__(256) void final_softplus_kernel(
    const float* __restrict__ node, const float* __restrict__ acc,
    float* __restrict__ out, int n) {
    int i = blockIdx.x * blockDim.x + threadIdx.x;
    if (i < n) out[i] = softplus_f(node[i] + acc[i]);
}

extern "C" void kernel_launch(void* const* d_in, const int* in_sizes, int n_in,
                              void* d_out, int out_size, void* d_ws, size_t ws_size,
                              hipStream_t stream) {
    (void)n_in; (void)out_size; (void)ws_size;
    const float* node_feats = (const float*)d_in[0];
    const float* edge_feats = (const float*)d_in[1];
    const int*   src        = (const int*)d_in[2];
    const int*   dst        = (const int*)d_in[3];
    const float* ws_w1 = (const float*)d_in[4];  const float* ws_b1 = (const float*)d_in[5];
    const float* ws_w2 = (const float*)d_in[6];  const float* ws_b2 = (const float*)d_in[7];
    const float* wd_w1 = (const float*)d_in[8];  const float* wd_b1 = (const float*)d_in[9];
    const float* wd_w2 = (const float*)d_in[10]; const float* wd_b2 = (const float*)d_in[11];
    const float* we_w1 = (const float*)d_in[12]; const float* we_b1 = (const float*)d_in[13];
    const float* we_w2 = (const float*)d_in[14]; const float* we_b2 = (const float*)d_in[15];
    const float* wm_w1 = (const float*)d_in[16]; const float* wm_b1 = (const float*)d_in[17];
    const float* wm_w2 = (const float*)d_in[18]; const float* wm_b2 = (const float*)d_in[19];

    const int n_nodes = in_sizes[0] / 64;
    const int n_edges = in_sizes[1] / 64;

    // workspace layout (floats)
    float* ws      = (float*)d_ws;
    float* hws_buf = ws;
    float* hwd_buf = ws + (size_t)n_nodes * 64;
    float* acc_buf = ws + 2 * (size_t)n_nodes * 64;
    float* packed  = ws + 3 * (size_t)n_nodes * 64;   // 8 layers x 4096 floats

    // 1) pack weight fragments (layers: ws1, ws2, wd1, wd2, we1, we2, wm1, wm2)
    pack_weights_kernel<<<dim3(16, 8), 256, 0, stream>>>(
        ws_w1, ws_w2, wd_w1, wd_w2, we_w1, we_w2, wm_w1, wm_w2, packed);

    // 2) zero scatter accumulator (fresh each call; capture-safe)
    hipMemsetAsync(acc_buf, 0, (size_t)n_nodes * 64 * sizeof(float), stream);

    // 3) node MLPs
    const int ntiles = (n_nodes + 15) / 16;
    const int nblocks = (ntiles + WAVES_PER_BLOCK - 1) / WAVES_PER_BLOCK;
    node_mlp_kernel<<<nblocks, 256, 0, stream>>>(
        node_feats, packed + 0 * 4096, packed + 1 * 4096, ws_b1, ws_b2, hws_buf, n_nodes);
    node_mlp_kernel<<<nblocks, 256, 0, stream>>>(
        node_feats, packed + 2 * 4096, packed + 3 * 4096, wd_b1, wd_b2, hwd_buf, n_nodes);

    // 4) fused edge pipeline + scatter
    const int etiles = (n_edges + 15) / 16;
    const int eblocks = (etiles + WAVES_PER_BLOCK - 1) / WAVES_PER_BLOCK;
    edge_pipeline_kernel<<<eblocks, 256, 0, stream>>>(
        edge_feats, src, dst, hws_buf, hwd_buf,
        packed + 4 * 4096, packed + 5 * 4096, we_b1, we_b2,
        packed + 6 * 4096, packed + 7 * 4096, wm_b1, wm_b2,
        acc_buf, n_edges);

    // 5) final softplus
    const int total = n_nodes * 64;
    final_softplus_kernel<<<(total + 255) / 256, 256, 0, stream>>>(
        node_feats, acc_buf, (float*)d_out, total);
}